// loss_point_27066883899873
// MI455X (gfx1250) — compile-verified
//
#include <hip/hip_runtime.h>
#include <hip/hip_bf16.h>
#include <math.h>

typedef __attribute__((ext_vector_type(2))) float v2f;
typedef __attribute__((ext_vector_type(8))) float v8f;

// ---------------- problem constants ----------------
#define N_IMG   8
#define L_PER   6
#define N_LANE  (N_IMG * L_PER)      // 48
#define RAW_P   20
#define DENSE   50
#define SPARSE  30
#define PPI     (L_PER * DENSE)      // 300 points per image
#define PPI_PAD 304                  // 19 WMMA n-tiles of 16
#define N_TILE  (PPI_PAD / 16)       // 19
#define N_BOX   25000
#define Wf      800.0f
#define Hf      320.0f
#define Rf      8.0f
#define CLIP_D  18.0f

// ---------------- output layout (floats, return order) ----------------
#define OFF_CONF 0                                   // 200000
#define OFF_OFF  (OFF_CONF + N_IMG * N_BOX)          // 400000
#define OFF_TPS  (OFF_OFF  + 2 * N_IMG * N_BOX)      // 2880
#define OFF_IDT  (OFF_TPS  + N_LANE * SPARSE * 2)    // 8000
#define OFF_CLS  (OFF_IDT  + N_IMG * 20 * 50)        // 200000
#define OFF_BRK  (OFF_CLS  + N_IMG * N_BOX)          // 200000

// ---------------- workspace layout (floats) ----------------
#define WS_PTX   0
#define WS_PTY   (WS_PTX  + N_IMG * PPI_PAD)
#define WS_PTPP  (WS_PTY  + N_IMG * PPI_PAD)
#define WS_PTCLS (WS_PTPP + N_IMG * PPI_PAD)
#define WS_PTBRK (WS_PTCLS+ N_IMG * PPI_PAD)
#define WS_PIXX  (WS_PTBRK+ N_IMG * PPI_PAD)
#define WS_PIXY  (WS_PIXX + N_IMG * PPI)

// =====================================================================
// Kernel 1: lane resampling + per-point channels + staging tables
// =====================================================================
__global__ void lane_prep_kernel(const float* __restrict__ lane_x,
                                 const float* __restrict__ lane_y,
                                 const int*   __restrict__ lane_cls,
                                 const float* __restrict__ bp_y,
                                 float* __restrict__ out,
                                 float* __restrict__ ws) {
    const int l = threadIdx.x;

    // pad points 300..303 of every image (lose every argmin, contribute 0)
    if (l < N_IMG * 4) {
        int img = l >> 2;
        int o = img * PPI_PAD + PPI + (l & 3);
        ws[WS_PTX  + o] = 0.0f;
        ws[WS_PTY  + o] = 0.0f;
        ws[WS_PTPP + o] = 1.0e30f;
        ws[WS_PTCLS+ o] = 0.0f;
        ws[WS_PTBRK+ o] = 0.0f;
    }

    if (l >= N_LANE) return;

    float lx[RAW_P], ly[RAW_P];
    #pragma unroll
    for (int k = 0; k < RAW_P; ++k) {
        lx[k] = lane_x[l * RAW_P + k];
        ly[k] = lane_y[l * RAW_P + k];
    }

    // sparse 30-point resample -> target_points_sparse
    for (int j = 0; j < SPARSE; ++j) {
        float pos = (float)j * (19.0f / 29.0f);
        int i0 = (int)floorf(pos); if (i0 > RAW_P - 2) i0 = RAW_P - 2;
        float fr = pos - (float)i0;
        float xs = lx[i0] * (1.0f - fr) + lx[i0 + 1] * fr;
        float ys = ly[i0] * (1.0f - fr) + ly[i0 + 1] * fr;
        out[OFF_TPS + (l * SPARSE + j) * 2 + 0] = xs;
        out[OFF_TPS + (l * SPARSE + j) * 2 + 1] = ys;
    }

    const int   cls  = lane_cls[l];
    const float fch  = (float)(cls / 10);
    const float bch  = (float)(cls % 10);
    const bool  isbp = (fch != bch);
    const float bpy  = bp_y[l];
    const float thr  = 5.0f / Hf;
    const int   img  = l / L_PER;
    const int   slot = (l % L_PER) * DENSE;

    for (int j = 0; j < DENSE; ++j) {
        float pos = (float)j * (19.0f / 49.0f);
        int i0 = (int)floorf(pos); if (i0 > RAW_P - 2) i0 = RAW_P - 2;
        float fr = pos - (float)i0;
        float x = lx[i0] * (1.0f - fr) + lx[i0 + 1] * fr;
        float y = ly[i0] * (1.0f - fr) + ly[i0 + 1] * fr;

        bool bigger  = y > bpy + thr;
        bool smaller = y < bpy - thr;
        float ch3bp  = bigger ? fch : (smaller ? bch : (j == 0 ? bpy : 0.0f));
        float ch3    = isbp ? ch3bp : fch;
        float ch2    = (isbp && !bigger && !smaller) ? 1.0f : 0.0f;

        int p  = slot + j;
        int o  = img * PPI_PAD + p;
        float sx = x * Wf, sy = y * Hf;
        ws[WS_PTX  + o] = sx;
        ws[WS_PTY  + o] = sy;
        ws[WS_PTPP + o] = sx * sx + sy * sy;
        ws[WS_PTCLS+ o] = ch3;
        ws[WS_PTBRK+ o] = ch2;

        int o2 = img * PPI + p;
        float fx = floorf(sx); fx = fx < 0.0f ? 0.0f : (fx > Wf ? Wf : fx);
        float fy = floorf(sy); fy = fy < 0.0f ? 0.0f : (fy > Hf ? Hf : fy);
        ws[WS_PIXX + o2] = fx;
        ws[WS_PIXY + o2] = fy;
    }
}

// =====================================================================
// Kernel 2: IDT mask, grid (4, 8) x 256 threads
// =====================================================================
__global__ void idt_kernel(const float* __restrict__ ws,
                           float* __restrict__ out) {
    __shared__ float sx[PPI], sy[PPI];
    const int img = blockIdx.y;
    for (int p = threadIdx.x; p < PPI; p += blockDim.x) {
        sx[p] = ws[WS_PIXX + img * PPI + p];
        sy[p] = ws[WS_PIXY + img * PPI + p];
    }
    __syncthreads();

    int cell = blockIdx.x * blockDim.x + threadIdx.x;
    if (cell >= 1000) return;                  // 20 x 50 grid
    int gyi = cell / 50, gxi = cell % 50;
    float gx = ((float)gxi + 0.5f) * 16.0f - 0.5f;
    float gy = ((float)gyi + 0.5f) * 16.0f - 0.5f;

    float mind = 3.0e38f;
    #pragma unroll 4
    for (int p = 0; p < PPI; ++p) {
        float dx = sx[p] - gx, dy = sy[p] - gy;
        mind = fminf(mind, dx * dx + dy * dy);
    }
    float d = sqrtf(mind);
    d = d > CLIP_D ? CLIP_D : d;
    out[OFF_IDT + img * 1000 + cell] = CLIP_D - d;
}

// =====================================================================
// Kernel 3: nearest-point assignment via V_WMMA_F32_16X16X4_F32
//   score[b][p] = |p|^2 - 2 c.p   (argmin-equivalent to squared distance)
//   A[16x4] = (cx, cy, 1, 0)  B[4x16] = (-2px; -2py; |p|^2; 0)
// B operand is pre-packed per (tile, lane) so the hot loop is one
// non-divergent ds_load_b64 + one WMMA per tile, fully unrolled.
// grid (391, 8) x 128 threads (4 waves, 16 boxes/wave)
// =====================================================================
__global__ void box_assign_kernel(const float* __restrict__ boxes,
                                  const float* __restrict__ ws,
                                  float* __restrict__ out) {
    __shared__ float s_bpk[N_TILE * 32 * 2];   // packed B VGPR image per lane
    __shared__ float s_px[PPI_PAD], s_py[PPI_PAD],
                     s_cls[PPI_PAD], s_brk[PPI_PAD];

    const int img = blockIdx.y;

    for (int p = threadIdx.x; p < PPI_PAD; p += blockDim.x) {
        int o = img * PPI_PAD + p;
        s_px[p]  = ws[WS_PTX  + o];
        s_py[p]  = ws[WS_PTY  + o];
        s_cls[p] = ws[WS_PTCLS+ o];
        s_brk[p] = ws[WS_PTBRK+ o];
    }
    // Pack B operand: entry e = (tile t, lane ln). Low lanes carry rows
    // K=0/K=1 (-2px, -2py); high lanes carry rows K=2/K=3 (|p|^2, 0).
    for (int e = threadIdx.x; e < N_TILE * 32; e += blockDim.x) {
        int t  = e >> 5;
        int ln = e & 31;
        int p  = t * 16 + (ln & 15);
        int o  = img * PPI_PAD + p;
        float bx, by;
        if (ln < 16) { bx = -2.0f * ws[WS_PTX + o]; by = -2.0f * ws[WS_PTY + o]; }
        else         { bx = ws[WS_PTPP + o];        by = 0.0f; }
        s_bpk[2 * e + 0] = bx;
        s_bpk[2 * e + 1] = by;
    }
    __syncthreads();

    const int lane    = threadIdx.x & 31;
    const int wave    = threadIdx.x >> 5;
    const int boxBase = blockIdx.x * 64 + wave * 16;
    const int m       = lane & 15;
    const int myBox   = boxBase + m;

    // A-matrix (16x4 f32): lanes 0-15 -> K=0,1 ; lanes 16-31 -> K=2,3
    float cx = 0.0f, cy = 0.0f;
    if (lane < 16 && myBox < N_BOX) {
        const float4 bx =
            reinterpret_cast<const float4*>(boxes)[img * N_BOX + myBox];
        cx = (bx.x + bx.z) * 0.5f * Wf;
        cy = (bx.y + bx.w) * 0.5f * Hf;
    }
    v2f A;
    A.x = (lane < 16) ? cx : 1.0f;   // K=0 | K=2
    A.y = (lane < 16) ? cy : 0.0f;   // K=1 | K=3

    float minv[8];
    int   mini[8];
    #pragma unroll
    for (int j = 0; j < 8; ++j) { minv[j] = 3.0e38f; mini[j] = 0; }

    const int n16 = lane & 15;
    const v2f* bpk = reinterpret_cast<const v2f*>(s_bpk);

    #pragma unroll
    for (int t = 0; t < N_TILE; ++t) {
        v2f B = bpk[t * 32 + lane];            // single ds_load_b64, uniform

        v8f c = {};
        c = __builtin_amdgcn_wmma_f32_16x16x4_f32(
                false, A, false, B, (short)0, c, false, false);

        // D layout: vgpr j, lanes 0-15 -> box j, lanes 16-31 -> box j+8;
        // point index = t*16 + (lane & 15) in both halves.
        int p = t * 16 + n16;
        #pragma unroll
        for (int j = 0; j < 8; ++j) {
            float s = c[j];
            if (s < minv[j]) { minv[j] = s; mini[j] = p; }
        }
    }

    // cross-lane argmin within each 16-lane half, then capture per-box index
    int myIdx = 1;
    #pragma unroll
    for (int j = 0; j < 8; ++j) {
        float v = minv[j];
        int   i = mini[j];
        #pragma unroll
        for (int off = 1; off < 16; off <<= 1) {
            float ov = __shfl_xor(v, off, 32);
            int   oi = __shfl_xor(i, off, 32);
            if (ov < v || (ov == v && oi < i)) { v = ov; i = oi; }
        }
        int iLow  = __shfl(i, 0, 32);   // min for box boxBase + j
        int iHigh = __shfl(i, 16, 32);  // min for box boxBase + j + 8
        if (lane == j)     myIdx = iLow;
        if (lane == j + 8) myIdx = iHigh;
    }

    if (lane < 16 && myBox < N_BOX) {
        int idx = myIdx;
        idx = idx < 1 ? 1 : (idx > PPI - 3 ? PPI - 3 : idx);   // clip [1,297]
        float px = s_px[idx], py = s_py[idx];
        float conf = (fabsf(px - cx) < Rf && fabsf(py - cy) < Rf) ? 1.0f : 0.0f;
        int gb = img * N_BOX + myBox;
        out[OFF_CONF + gb]        = conf;
        out[OFF_OFF + 2 * gb + 0] = (px - (cx - Rf)) * (1.0f / (2.0f * Rf));
        out[OFF_OFF + 2 * gb + 1] = (py - (cy - Rf)) * (1.0f / (2.0f * Rf));
        out[OFF_CLS + gb]         = s_cls[idx];
        out[OFF_BRK + gb]         = s_brk[idx];
    }
}

// =====================================================================
extern "C" void kernel_launch(void* const* d_in, const int* in_sizes, int n_in,
                              void* d_out, int out_size, void* d_ws, size_t ws_size,
                              hipStream_t stream) {
    const float* lane_x   = (const float*)d_in[0];
    const float* lane_y   = (const float*)d_in[1];
    const int*   lane_cls = (const int*)  d_in[2];
    const float* bp_y     = (const float*)d_in[3];
    const float* boxes    = (const float*)d_in[4];
    float* out = (float*)d_out;
    float* ws  = (float*)d_ws;

    lane_prep_kernel<<<1, 64, 0, stream>>>(lane_x, lane_y, lane_cls, bp_y, out, ws);

    idt_kernel<<<dim3(4, N_IMG), 256, 0, stream>>>(ws, out);

    // 64 boxes per block -> ceil(25000/64) = 391 blocks per image
    box_assign_kernel<<<dim3((N_BOX + 63) / 64, N_IMG), 128, 0, stream>>>(boxes, ws, out);
}